// Item_GraphConvolution_mid_attention_65609920414006
// MI455X (gfx1250) — compile-verified
//
#include <hip/hip_runtime.h>

#define NN 50000      // nodes
#define NE 800000     // edges
#define FIN 256       // input features
#define DD 256        // embedding size

typedef __attribute__((ext_vector_type(2))) float v2f;
typedef __attribute__((ext_vector_type(4))) float v4f;
typedef __attribute__((ext_vector_type(8))) float v8f;

// -------------------------------------------------------------------------
// Dense GEMM + ReLU using fp32 WMMA (v_wmma_f32_16x16x4_f32).
// Block = 128 threads = 4 waves. blockIdx.x selects a 16-row strip of the
// output; wave w owns columns [w*64, w*64+64). K walked in steps of 4.
// VGPR layouts per CDNA5 ISA 7.12.2 (A 16x4 split across lane halves by K,
// B 4x16 mirrored, C/D v8f rows r / r+8 by lane half).
// -------------------------------------------------------------------------
__global__ __launch_bounds__(128) void gemm_relu_wmma(
    const float* __restrict__ feat,     // [NN, FIN] row-major
    const float* __restrict__ W,        // [FIN, DD] row-major
    float* __restrict__ support) {      // [NN, DD]
  const int wave = threadIdx.x >> 5;
  const int lane = threadIdx.x & 31;
  const int half = lane >> 4;           // 0 or 1
  const int lm   = lane & 15;
  const int m0 = blockIdx.x << 4;       // 16-row strip
  const int n0 = wave << 6;             // 64-col slab per wave

  v8f acc0 = {}, acc1 = {}, acc2 = {}, acc3 = {};

  const float* arow = feat + (size_t)(m0 + lm) * FIN + 2 * half;
  const float* bcol = W + n0 + lm;

  for (int k0 = 0; k0 < FIN; k0 += 4) {
    v2f a;
    a.x = arow[k0];       // K = k0 + 2*half
    a.y = arow[k0 + 1];   // K = k0 + 2*half + 1

    const float* bk0 = bcol + (size_t)(k0 + 2 * half) * DD;
    const float* bk1 = bk0 + DD;
    v2f b0, b1, b2, b3;
    b0.x = bk0[0];  b0.y = bk1[0];
    b1.x = bk0[16]; b1.y = bk1[16];
    b2.x = bk0[32]; b2.y = bk1[32];
    b3.x = bk0[48]; b3.y = bk1[48];

    acc0 = __builtin_amdgcn_wmma_f32_16x16x4_f32(false, a, false, b0, (short)0, acc0, false, false);
    acc1 = __builtin_amdgcn_wmma_f32_16x16x4_f32(false, a, false, b1, (short)0, acc1, false, false);
    acc2 = __builtin_amdgcn_wmma_f32_16x16x4_f32(false, a, false, b2, (short)0, acc2, false, false);
    acc3 = __builtin_amdgcn_wmma_f32_16x16x4_f32(false, a, false, b3, (short)0, acc3, false, false);
  }

  float* srow = support + (size_t)m0 * DD + n0 + lm;
#pragma unroll
  for (int r = 0; r < 8; ++r) {
    const int m = r + (half << 3);
    float* p = srow + (size_t)m * DD;
    float x0 = acc0[r]; p[0]  = x0 > 0.f ? x0 : 0.f;
    float x1 = acc1[r]; p[16] = x1 > 0.f ? x1 : 0.f;
    float x2 = acc2[r]; p[32] = x2 > 0.f ? x2 : 0.f;
    float x3 = acc3[r]; p[48] = x3 > 0.f ? x3 : 0.f;
  }
}

// ------------------------- CSR build (every call) -------------------------

__global__ __launch_bounds__(256) void zero_i32(int* __restrict__ p, int n) {
  int i = blockIdx.x * 256 + threadIdx.x;
  if (i < n) p[i] = 0;
}

__global__ __launch_bounds__(256) void degree_hist(const int* __restrict__ erow,
                                                   int* __restrict__ deg) {
  int e = blockIdx.x * 256 + threadIdx.x;
  if (e < NE) atomicAdd(&deg[erow[e]], 1);
}

// Single-block Hillis-Steele scan over NN degrees -> exclusive rowptr
// (+ cursor copy for the scatter pass, + rowptr[NN] = E).
#define SCAN_T 1024
__global__ __launch_bounds__(SCAN_T) void scan_rowptr(const int* __restrict__ deg,
                                                      int* __restrict__ rowptr,
                                                      int* __restrict__ cursor) {
  __shared__ int sm[SCAN_T];
  __shared__ int running;
  if (threadIdx.x == 0) running = 0;
  __syncthreads();
  for (int base = 0; base < NN; base += SCAN_T) {
    const int i = base + (int)threadIdx.x;
    const int v = (i < NN) ? deg[i] : 0;
    sm[threadIdx.x] = v;
    __syncthreads();
    for (int off = 1; off < SCAN_T; off <<= 1) {
      int t = (threadIdx.x >= (unsigned)off) ? sm[threadIdx.x - off] : 0;
      __syncthreads();
      sm[threadIdx.x] += t;
      __syncthreads();
    }
    const int incl = sm[threadIdx.x];        // inclusive scan of this chunk
    const int base_run = running;            // stable (synced writes)
    const int excl = base_run + incl - v;
    if (i < NN) { rowptr[i] = excl; cursor[i] = excl; }
    __syncthreads();                         // everyone has read `running`
    if (threadIdx.x == SCAN_T - 1) running = base_run + incl;
    __syncthreads();
  }
  if (threadIdx.x == 0) rowptr[NN] = running;   // == NE
}

__global__ __launch_bounds__(256) void scatter_edges(const int* __restrict__ erow,
                                                     int* __restrict__ cursor,
                                                     int* __restrict__ eidx) {
  int e = blockIdx.x * 256 + threadIdx.x;
  if (e >= NE) return;
  int pos = atomicAdd(&cursor[erow[e]], 1);
  eidx[pos] = e;
}

// ----------------------- gather-form SpMM + epilogues ---------------------
// One wave per destination row; lane l owns cols [4l,4l+4) and [128+4l,128+4l+4).
// Accumulation entirely in registers -> zero atomics. Gathers of x[col] are
// coalesced v4f loads served from the 192MB L2 (support/ax stay resident).
// Final outputs are write-once -> non-temporal stores so they bypass/rinse L2.

__global__ __launch_bounds__(256) void spmm_low(
    const int* __restrict__ rowptr, const int* __restrict__ eidx,
    const int* __restrict__ ecol, const float* __restrict__ evals,
    const float* __restrict__ sup,      // x = support
    float* __restrict__ ax,             // A @ support (ws, L2-resident)
    float* __restrict__ out) {          // d_out base
  const int wave = threadIdx.x >> 5;
  const int lane = threadIdx.x & 31;
  const int n = blockIdx.x * 8 + wave;
  if (n >= NN) return;

  const int beg = rowptr[n], end = rowptr[n + 1];
  const int j0 = lane << 2, j1 = j0 + 128;

  v4f a0 = {}, a1 = {};
  for (int p = beg; p < end; ++p) {
    const int e = eidx[p];
    const float v = evals[e];
    const float* xr = sup + ((size_t)ecol[e] << 8);
    a0 += v * *(const v4f*)(xr + j0);
    a1 += v * *(const v4f*)(xr + j1);
  }

  const size_t rb = (size_t)n << 8;
  *(v4f*)(ax + rb + j0) = a0;
  *(v4f*)(ax + rb + j1) = a1;

  const v4f s0 = *(const v4f*)(sup + rb + j0);
  const v4f s1 = *(const v4f*)(sup + rb + j1);
  const v4f l0 = a0 + s0, l1 = a1 + s1;

  const size_t total = (size_t)NN * DD;
  __builtin_nontemporal_store(l0, (v4f*)(out + 2 * total + rb + j0));  // output_low
  __builtin_nontemporal_store(l1, (v4f*)(out + 2 * total + rb + j1));
  __builtin_nontemporal_store(l0, (v4f*)(out + ((size_t)n << 9) + j0)); // output[n,0,:]
  __builtin_nontemporal_store(l1, (v4f*)(out + ((size_t)n << 9) + j1));
}

__global__ __launch_bounds__(256) void spmm_mid(
    const int* __restrict__ rowptr, const int* __restrict__ eidx,
    const int* __restrict__ ecol, const float* __restrict__ evals,
    const float* __restrict__ ax,       // x = A @ support
    const float* __restrict__ sup,
    float* __restrict__ out) {          // d_out base
  const int wave = threadIdx.x >> 5;
  const int lane = threadIdx.x & 31;
  const int n = blockIdx.x * 8 + wave;
  if (n >= NN) return;

  const int beg = rowptr[n], end = rowptr[n + 1];
  const int j0 = lane << 2, j1 = j0 + 128;

  v4f a0 = {}, a1 = {};
  for (int p = beg; p < end; ++p) {
    const int e = eidx[p];
    const float v = evals[e];
    const float* xr = ax + ((size_t)ecol[e] << 8);
    a0 += v * *(const v4f*)(xr + j0);
    a1 += v * *(const v4f*)(xr + j1);
  }

  const size_t rb = (size_t)n << 8;
  const v4f s0 = *(const v4f*)(sup + rb + j0);
  const v4f s1 = *(const v4f*)(sup + rb + j1);
  const v4f m0 = a0 - s0, m1 = a1 - s1;

  const size_t total = (size_t)NN * DD;
  __builtin_nontemporal_store(m0, (v4f*)(out + 3 * total + rb + j0));  // output_mid
  __builtin_nontemporal_store(m1, (v4f*)(out + 3 * total + rb + j1));
  __builtin_nontemporal_store(m0, (v4f*)(out + ((size_t)n << 9) + DD + j0)); // output[n,1,:]
  __builtin_nontemporal_store(m1, (v4f*)(out + ((size_t)n << 9) + DD + j1));
}

extern "C" void kernel_launch(void* const* d_in, const int* in_sizes, int n_in,
                              void* d_out, int out_size, void* d_ws, size_t ws_size,
                              hipStream_t stream) {
  const float* feat  = (const float*)d_in[0];   // [NN, FIN]
  const float* W     = (const float*)d_in[1];   // [FIN, DD]
  const int*   erow  = (const int*)d_in[2];     // [NE]
  const int*   ecol  = (const int*)d_in[3];     // [NE]
  const float* evals = (const float*)d_in[4];   // [NE]
  float* out = (float*)d_out;                   // [NN*2*DD | NN*DD | NN*DD]

  const long long nd = (long long)NN * DD;

  float* support = (float*)d_ws;                // 51.2 MB
  float* ax      = support + nd;                // 51.2 MB
  int*   deg     = (int*)(ax + nd);             // NN  (also reused as cursor src)
  int*   rowptr  = deg + NN;                    // NN + 1
  int*   cursor  = rowptr + NN + 1;             // NN
  int*   eidx    = cursor + NN;                 // NE

  const int edgeBlocks = (NE + 255) / 256;      // 3125
  const int nodeBlocks = (NN + 255) / 256;      // 196
  const int gatherBlocks = (NN + 7) / 8;        // 6250 (8 waves/block, 1 row/wave)

  // CSR build (must rerun every call; ~5 MB traffic, negligible)
  zero_i32<<<nodeBlocks, 256, 0, stream>>>(deg, NN);
  degree_hist<<<edgeBlocks, 256, 0, stream>>>(erow, deg);
  scan_rowptr<<<1, SCAN_T, 0, stream>>>(deg, rowptr, cursor);
  scatter_edges<<<edgeBlocks, 256, 0, stream>>>(erow, cursor, eidx);

  // support = relu(feature @ W)   (fp32 WMMA)
  gemm_relu_wmma<<<NN / 16, 128, 0, stream>>>(feat, W, support);

  // ax = A@support ; output_low slab + output[:,0,:]
  spmm_low<<<gatherBlocks, 256, 0, stream>>>(rowptr, eidx, ecol, evals,
                                             support, ax, out);
  // output_mid = A@ax - support ; output[:,1,:]
  spmm_mid<<<gatherBlocks, 256, 0, stream>>>(rowptr, eidx, ecol, evals,
                                             ax, support, out);
}